// OCGAT_51616916963799
// MI455X (gfx1250) — compile-verified
//
#include <hip/hip_runtime.h>
#include <hip/hip_bf16.h>
#include <math.h>

// ---------------- problem constants (match reference) ----------------
#define NN    20000     // nodes (divisible by 32)
#define EE    320000    // raw edges
#define ETOT  (EE + NN) // edges + self loops
#define INC   128       // in channels
#define HH    8         // heads
#define CC    33        // per-head dim
#define DD    (HH * CC) // 264 logical width
#define DP    272       // padded width (17 x 16)
#define OUTD  (DD / 2)  // 132
#define OUTP  144       // padded readout width (9 x 16)
#define NEG_SLOPE 0.2f

typedef float v2f __attribute__((ext_vector_type(2)));
typedef float v8f __attribute__((ext_vector_type(8)));

// ---------------- generic fill ----------------
__global__ void gat_fill(float* __restrict__ p, float v, int n) {
    int i = blockIdx.x * blockDim.x + threadIdx.x;
    if (i < n) p[i] = v;
}

// ---------------- transpose + zero-pad weights: Wt[col][k] ----------------
// W is [K, Ncols] row-major; Wt is [Npad, K] row-major, cols >= Ncols are 0.
__global__ void gat_wprep(const float* __restrict__ W, float* __restrict__ Wt,
                          int K, int Ncols, int Npad) {
    int idx = blockIdx.x * blockDim.x + threadIdx.x;
    if (idx >= Npad * K) return;
    int col = idx / K;
    int k   = idx - col * K;
    Wt[idx] = (col < Ncols) ? W[k * Ncols + col] : 0.f;
}

// ---------------- WMMA f32 GEMM ----------------
// Out[M, ldo] = A[M, lda(:K)] @ Wt^T  (+ bias), Wt is [Npad][K].
// One wave32 computes a 32x16 output tile: 2 accumulators sharing the B fragment.
// M % 32 == 0, K % 4 == 0, Npad % 16 == 0  -> no load guards, EXEC all-ones.
__global__ void gat_gemm_wmma(const float* __restrict__ A, int lda,
                              const float* __restrict__ Wt,
                              const float* __restrict__ bias, // may be null
                              float* __restrict__ Out, int ldo,
                              int M, int K, int Npad, int Nvalid) {
    const int lane   = threadIdx.x & 31;
    const int waveId = (blockIdx.x * blockDim.x + threadIdx.x) >> 5;
    const int tilesN = Npad >> 4;
    const int tilesM = M >> 5;
    const int tm = waveId / tilesN;
    const int tn = waveId % tilesN;
    if (tm >= tilesM) return;  // wave-uniform

    const int halfSel = lane >> 4;      // which K-pair this half-wave holds
    const int lane16  = lane & 15;
    const int row0 = tm * 32 + lane16;  // rows 0..15 of tile
    const int col  = tn * 16 + lane16;

    const float* Arow0 = A  + (size_t)row0 * lda;
    const float* Arow1 = Arow0 + (size_t)16 * lda;
    const float* Wcol  = Wt + (size_t)col * K;
    const int koff = halfSel * 2;

    v8f acc0 = {0.f, 0.f, 0.f, 0.f, 0.f, 0.f, 0.f, 0.f};
    v8f acc1 = acc0;

    #pragma unroll 2
    for (int k = 0; k < K; k += 4) {
        const int ka = k + koff;
        v2f a0 = *(const v2f*)(Arow0 + ka);   // global_load_b64
        v2f a1 = *(const v2f*)(Arow1 + ka);   // global_load_b64
        v2f b  = *(const v2f*)(Wcol  + ka);   // global_load_b64
        acc0 = __builtin_amdgcn_wmma_f32_16x16x4_f32(
            false, a0, false, b, (short)0, acc0, false, false);
        acc1 = __builtin_amdgcn_wmma_f32_16x16x4_f32(
            false, a1, false, b, (short)0, acc1, false, false);
    }

    if (col < Nvalid) {
        const float badd = bias ? bias[col] : 0.f;
        const int rbase = tm * 32 + halfSel * 8;
        #pragma unroll
        for (int r = 0; r < 8; ++r) {
            Out[(size_t)(rbase + r)      * ldo + col] = acc0[r] + badd;
            Out[(size_t)(rbase + 16 + r) * ldo + col] = acc1[r] + badd;
        }
    }
}

// ---------------- per-node attention logits ----------------
__global__ void gat_alpha(const float* __restrict__ xh,
                          const float* __restrict__ a_s,
                          const float* __restrict__ a_d,
                          float* __restrict__ alpha_s,
                          float* __restrict__ alpha_d) {
    int idx = blockIdx.x * blockDim.x + threadIdx.x;
    if (idx >= NN * HH) return;
    int n = idx >> 3;
    int h = idx & 7;
    const float* row = xh + (size_t)n * DP + h * CC;
    const float* vs  = a_s + h * CC;
    const float* vd  = a_d + h * CC;
    float ss = 0.f, sd = 0.f;
    #pragma unroll
    for (int c = 0; c < CC; ++c) {
        float v = row[c];
        ss += v * vs[c];
        sd += v * vd[c];
    }
    alpha_s[idx] = ss;
    alpha_d[idx] = sd;
}

__device__ __forceinline__ int   edge_src(const int* ei, int e) { return (e < EE) ? ei[e]      : (e - EE); }
__device__ __forceinline__ int   edge_dst(const int* ei, int e) { return (e < EE) ? ei[EE + e] : (e - EE); }
__device__ __forceinline__ float lrelu(float v) { return v >= 0.f ? v : NEG_SLOPE * v; }

__device__ __forceinline__ void atomicMaxF32(float* addr, float val) {
    if (val >= 0.f) atomicMax((int*)addr, __float_as_int(val));
    else            atomicMin((unsigned int*)addr, __float_as_uint(val));
}

// ---------------- segment max over destinations ----------------
__global__ void gat_edge_max(const int* __restrict__ ei,
                             const float* __restrict__ alpha_s,
                             const float* __restrict__ alpha_d,
                             float* __restrict__ m) {
    int idx = blockIdx.x * blockDim.x + threadIdx.x;
    if (idx >= ETOT * HH) return;
    int e = idx >> 3, h = idx & 7;
    int s = edge_src(ei, e), d = edge_dst(ei, e);
    float v = lrelu(alpha_s[s * HH + h] + alpha_d[d * HH + h]);
    atomicMaxF32(&m[d * HH + h], v);
}

// ---------------- segment sum of exp ----------------
__global__ void gat_edge_sum(const int* __restrict__ ei,
                             const float* __restrict__ alpha_s,
                             const float* __restrict__ alpha_d,
                             const float* __restrict__ m,
                             float* __restrict__ denom) {
    int idx = blockIdx.x * blockDim.x + threadIdx.x;
    if (idx >= ETOT * HH) return;
    int e = idx >> 3, h = idx & 7;
    int s = edge_src(ei, e), d = edge_dst(ei, e);
    float v  = lrelu(alpha_s[s * HH + h] + alpha_d[d * HH + h]);
    float ex = expf(v - m[d * HH + h]);
    atomicAdd(&denom[d * HH + h], ex);
}

// ---------------- per-edge-head softmax coefficient ----------------
__global__ void gat_coef(const int* __restrict__ ei,
                         const float* __restrict__ alpha_s,
                         const float* __restrict__ alpha_d,
                         const float* __restrict__ m,
                         const float* __restrict__ denom,
                         float* __restrict__ coef) {
    int idx = blockIdx.x * blockDim.x + threadIdx.x;
    if (idx >= ETOT * HH) return;
    int e = idx >> 3, h = idx & 7;
    int s = edge_src(ei, e), d = edge_dst(ei, e);
    float v  = lrelu(alpha_s[s * HH + h] + alpha_d[d * HH + h]);
    float ex = expf(v - m[d * HH + h]);
    coef[idx] = ex / (denom[d * HH + h] + 1e-16f);
}

// ---------------- message scatter: warp per edge ----------------
__global__ void gat_scatter(const int* __restrict__ ei,
                            const float* __restrict__ xh,
                            const float* __restrict__ coef,
                            float* __restrict__ out) {
    int warp = (blockIdx.x * blockDim.x + threadIdx.x) >> 5;
    int lane = threadIdx.x & 31;
    if (warp >= ETOT) return;
    int s = edge_src(ei, warp), d = edge_dst(ei, warp);
    const float* srow = xh + (size_t)s * DP;
    float*       drow = out + (size_t)d * DP;
    const float* crow = coef + (size_t)warp * HH;
    for (int c = lane; c < DD; c += 32) {
        int h = c / CC;
        atomicAdd(&drow[c], srow[c] * crow[h]);
    }
}

// ---------------- bias + ELU (logical cols only; pad cols stay 0) ----------------
__global__ void gat_bias_elu(float* __restrict__ h, const float* __restrict__ b) {
    int idx = blockIdx.x * blockDim.x + threadIdx.x;
    if (idx >= NN * DD) return;
    int n = idx / DD;
    int c = idx - n * DD;
    float* p = h + (size_t)n * DP + c;
    float v = *p + b[c];
    *p = v > 0.f ? v : (expf(v) - 1.f);
}

// ---------------- driver ----------------
extern "C" void kernel_launch(void* const* d_in, const int* in_sizes, int n_in,
                              void* d_out, int out_size, void* d_ws, size_t ws_size,
                              hipStream_t stream) {
    (void)in_sizes; (void)n_in; (void)out_size; (void)ws_size;

    const float* x  = (const float*)d_in[0];
    const int*   ei = (const int*)d_in[1];
    const float* Wl[4]  = {(const float*)d_in[2],  (const float*)d_in[6],
                           (const float*)d_in[10], (const float*)d_in[14]};
    const float* asl[4] = {(const float*)d_in[3],  (const float*)d_in[7],
                           (const float*)d_in[11], (const float*)d_in[15]};
    const float* adl[4] = {(const float*)d_in[4],  (const float*)d_in[8],
                           (const float*)d_in[12], (const float*)d_in[16]};
    const float* bl[4]  = {(const float*)d_in[5],  (const float*)d_in[9],
                           (const float*)d_in[13], (const float*)d_in[17]};
    const float* Whead = (const float*)d_in[18];
    const float* bhead = (const float*)d_in[19];
    float* out = (float*)d_out;

    // workspace layout (floats)
    float* bufA    = (float*)d_ws;                 // xh [N, DP]
    float* bufB    = bufA + (size_t)NN * DP;       // h  [N, DP]
    float* alpha_s = bufB + (size_t)NN * DP;       // [N, 8]
    float* alpha_d = alpha_s + NN * HH;
    float* mbuf    = alpha_d + NN * HH;
    float* denom   = mbuf + NN * HH;
    float* coef    = denom + NN * HH;              // [ETOT, 8]
    float* Wt      = coef + (size_t)ETOT * HH;     // [DP, 264] max

    const int T = 256;
    auto blocksFor = [&](long n) { return (int)((n + T - 1) / T); };
    auto gemmBlocks = [&](int npad) {
        long waves = (long)(NN / 32) * (npad / 16);
        return blocksFor(waves * 32);
    };
    const int nhBlocks = blocksFor((long)NN * HH);
    const int ehBlocks = blocksFor((long)ETOT * HH);
    const int scBlocks = blocksFor((long)ETOT * 32);
    const int ndBlocks = blocksFor((long)NN * DD);

    const float NEGINF = -__builtin_inff();

    for (int l = 0; l < 4; ++l) {
        const float* hin = (l == 0) ? x : bufB;
        const int    K   = (l == 0) ? INC : DD;
        const int    lda = (l == 0) ? INC : DP;
        // 0) transpose+pad weights -> Wt [DP][K]
        gat_wprep<<<blocksFor((long)DP * K), T, 0, stream>>>(Wl[l], Wt, K, DD, DP);
        // 1) xh = hin @ W_l  (WMMA, unguarded b64 loads)
        gat_gemm_wmma<<<gemmBlocks(DP), T, 0, stream>>>(hin, lda, Wt, nullptr,
                                                        bufA, DP, NN, K, DP, DP);
        // 2) attention logits
        gat_alpha<<<nhBlocks, T, 0, stream>>>(bufA, asl[l], adl[l], alpha_s, alpha_d);
        // 3) init segment buffers + output accumulator (incl. pad cols)
        gat_fill<<<nhBlocks, T, 0, stream>>>(mbuf, NEGINF, NN * HH);
        gat_fill<<<nhBlocks, T, 0, stream>>>(denom, 0.f, NN * HH);
        gat_fill<<<blocksFor((long)NN * DP), T, 0, stream>>>(bufB, 0.f, NN * DP);
        // 4) segment softmax over incoming edges
        gat_edge_max<<<ehBlocks, T, 0, stream>>>(ei, alpha_s, alpha_d, mbuf);
        gat_edge_sum<<<ehBlocks, T, 0, stream>>>(ei, alpha_s, alpha_d, mbuf, denom);
        gat_coef<<<ehBlocks, T, 0, stream>>>(ei, alpha_s, alpha_d, mbuf, denom, coef);
        // 5) aggregate messages
        gat_scatter<<<scBlocks, T, 0, stream>>>(ei, bufA, coef, bufB);
        // 6) bias + ELU
        gat_bias_elu<<<ndBlocks, T, 0, stream>>>(bufB, bl[l]);
    }

    // readout: out[N,132] = h @ W_head + b_head (store-guarded at col>=132)
    gat_wprep<<<blocksFor((long)OUTP * DD), T, 0, stream>>>(Whead, Wt, DD, OUTD, OUTP);
    gat_gemm_wmma<<<gemmBlocks(OUTP), T, 0, stream>>>(bufB, DP, Wt, bhead,
                                                      out, OUTD, NN, DD, OUTP, OUTD);
}